// TopKLayer_65343632441502
// MI455X (gfx1250) — compile-verified
//
#include <hip/hip_runtime.h>
#include <stdint.h>

// ---------------------------------------------------------------------------
// TopK (|value|) mask over the H*W axis, per (n, c) row.
//   x: [64, 256, 64, 64] fp32 -> rows of HW=4096 elements, keep k=409 by |x|.
// Memory-bound: 512 MB total traffic -> ~22us floor @ 23.3 TB/s.
// One workgroup per row; data staged via CDNA5 async global<->LDS ops
// (GLOBAL_LOAD_ASYNC_TO_LDS_B128 / GLOBAL_STORE_ASYNC_FROM_LDS_B128,
//  tracked with ASYNCcnt, drained with s_wait_asynccnt).
// ---------------------------------------------------------------------------

#define HW       4096
#define K_KEEP   409          // int(max(1, 0.1 * 4096))
#define THREADS  256          // 8 waves (wave32)
#define EPT      16           // elements per thread  (4096 / 256)
#define CPT      4            // 16-byte chunks per thread (1024 chunks / 256)

typedef int v4i __attribute__((vector_size(16)));
typedef __attribute__((address_space(1))) v4i g_v4i;   // global int4
typedef __attribute__((address_space(3))) v4i l_v4i;   // LDS int4

__device__ __forceinline__ void async_g2l_b128(const void* gaddr, void* laddr) {
#if __has_builtin(__builtin_amdgcn_global_load_async_to_lds_b128)
    __builtin_amdgcn_global_load_async_to_lds_b128(
        (g_v4i*)gaddr, (l_v4i*)laddr, /*offset=*/0, /*cpol=*/0);
#else
    unsigned long long ga = (unsigned long long)gaddr;
    unsigned la = (unsigned)(unsigned long long)(l_v4i*)laddr;
    asm volatile("global_load_async_to_lds_b128 %0, %1, off"
                 :: "v"(la), "v"(ga) : "memory");
#endif
}

__device__ __forceinline__ void async_l2g_b128(void* gaddr, void* laddr) {
#if __has_builtin(__builtin_amdgcn_global_store_async_from_lds_b128)
    __builtin_amdgcn_global_store_async_from_lds_b128(
        (g_v4i*)gaddr, (l_v4i*)laddr, /*offset=*/0, /*cpol=*/0);
#else
    unsigned long long ga = (unsigned long long)gaddr;
    unsigned la = (unsigned)(unsigned long long)(l_v4i*)laddr;
    asm volatile("global_store_async_from_lds_b128 %0, %1, off"
                 :: "v"(ga), "v"(la) : "memory");
#endif
}

__device__ __forceinline__ void wait_async0() {
#if __has_builtin(__builtin_amdgcn_s_wait_asynccnt)
    __builtin_amdgcn_s_wait_asynccnt(0);
#else
    asm volatile("s_wait_asynccnt 0x0" ::: "memory");
#endif
}

__global__ __launch_bounds__(THREADS) void topk_hw_mask_kernel(
        const float* __restrict__ x, float* __restrict__ out)
{
    __shared__ float    row[HW];   // 16 KB row staging
    __shared__ unsigned cnt[32];   // one counter per bit of the radix search

    const unsigned t    = threadIdx.x;
    const size_t   base = (size_t)blockIdx.x * (size_t)HW;

    if (t < 32) cnt[t] = 0u;

    // ---- 1) async copy global -> LDS (ASYNCcnt-tracked, bypasses VGPRs) ----
    const char* gsrc = (const char*)(x + base);
    char*       lrow = (char*)row;
#pragma unroll
    for (int j = 0; j < CPT; ++j) {
        const int boff = (int)(t + THREADS * j) * 16;   // disjoint 16B chunks
        async_g2l_b128(gsrc + boff, lrow + boff);
    }
    wait_async0();
    __syncthreads();

    // ---- 2) keys = |x| bits (monotone in |x|), conflict-free stride-256 ----
    unsigned key[EPT];
#pragma unroll
    for (int j = 0; j < EPT; ++j)
        key[j] = __float_as_uint(row[t + THREADS * j]) & 0x7fffffffu;

    // ---- 3) bitwise search for T = k-th largest key:
    //          max prefix with count(key >= prefix) >= K_KEEP ----
    unsigned prefix = 0u;
    for (int bit = 30; bit >= 0; --bit) {
        const unsigned cand = prefix | (1u << bit);
        unsigned c = 0u;
#pragma unroll
        for (int j = 0; j < EPT; ++j) c += (key[j] >= cand) ? 1u : 0u;
        // wave32 tree reduction
        c += __shfl_down(c, 16, 32);
        c += __shfl_down(c,  8, 32);
        c += __shfl_down(c,  4, 32);
        c += __shfl_down(c,  2, 32);
        c += __shfl_down(c,  1, 32);
        if ((t & 31u) == 0u) atomicAdd(&cnt[bit], c);   // 8 atomics / iter
        __syncthreads();
        if (cnt[bit] >= (unsigned)K_KEEP) prefix = cand;
    }

    // ---- 4) mask in LDS: zero everything below the threshold ----
#pragma unroll
    for (int j = 0; j < EPT; ++j)
        if (key[j] < prefix) row[t + THREADS * j] = 0.0f;
    __syncthreads();

    // ---- 5) async store LDS -> global ----
    char* gdst = (char*)(out + base);
#pragma unroll
    for (int j = 0; j < CPT; ++j) {
        const int boff = (int)(t + THREADS * j) * 16;
        async_l2g_b128(gdst + boff, lrow + boff);
    }
    wait_async0();   // (S_ENDPGM also waits idle, this makes it explicit)
}

extern "C" void kernel_launch(void* const* d_in, const int* in_sizes, int n_in,
                              void* d_out, int out_size, void* d_ws, size_t ws_size,
                              hipStream_t stream) {
    (void)n_in; (void)d_ws; (void)ws_size; (void)in_sizes;
    const float* x   = (const float*)d_in[0];
    float*       out = (float*)d_out;
    const int rows = out_size / HW;          // 64*256 = 16384 rows
    topk_hw_mask_kernel<<<rows, THREADS, 0, stream>>>(x, out);
}